// VisibilityRebuildModule_34488587387459
// MI455X (gfx1250) — compile-verified
//
#include <hip/hip_runtime.h>
#include <hip/hip_bf16.h>
#include <math.h>

typedef float v2f __attribute__((ext_vector_type(2)));
typedef float v4f __attribute__((ext_vector_type(4)));
typedef float v8f __attribute__((ext_vector_type(8)));

#define NPIX 65536   // 256*256
#define WH   256
#define NKPT 68

// Phase B launch geometry (kept in sync with kernel_launch below)
#define TB_BLOCKS_PER_BATCH 64
#define TB_THREADS          256
#define TB_WAVES_PER_BATCH  (TB_BLOCKS_PER_BATCH * (TB_THREADS / 32))   // 512
#define TILES_PER_BATCH     (NPIX / 16)                                  // 4096
#define TILES_PER_WAVE      (TILES_PER_BATCH / TB_WAVES_PER_BATCH)       // 8

// ---------------------------------------------------------------------------
// Phase A: per-batch keypoint gather + weighted Procrustes (tiny work).
// One block per batch. Threads 0..67 gather; thread 0 does the serial solve.
// Writes 12 floats per batch to Mout: Rs (row-major 3x3, = R*scale) then T.
// ---------------------------------------------------------------------------
__global__ __launch_bounds__(128) void vrm_procrustes_kernel(
    const float* __restrict__ Offset,
    const float* __restrict__ Posmap,
    const float* __restrict__ attention,
    const float* __restrict__ mean_posmap,
    const int*   __restrict__ uv,
    float* __restrict__ Mout)
{
    __shared__ float s_src[NKPT][3];
    __shared__ float s_dst[NKPT][3];
    __shared__ float s_w[NKPT];

    const int b = blockIdx.x;
    const int k = threadIdx.x;
    const float* pm  = Posmap + (size_t)b * 3 * NPIX;
    const float* off = Offset + (size_t)b * 3 * NPIX;

    if (k < NKPT) {
        int iu = uv[2 * k + 0];
        int iv = uv[2 * k + 1];
        int idx = iu * WH + iv;
        float dst[3], src[3];
        #pragma unroll
        for (int c = 0; c < 3; ++c) {
            dst[c] = pm[c * NPIX + idx];
            src[c] = off[c * NPIX + idx] + mean_posmap[c * NPIX + idx];
            s_dst[k][c] = dst[c];
            s_src[k][c] = src[c];
        }
        // visibility z = ab0*bc1 - ab1*bc0 at (iu, iv)
        float p0 = dst[0], p1 = dst[1];
        float d0 = 0.f, d1 = 0.f, r0 = 0.f, r1 = 0.f;
        if (iv >= 1) { d0 = pm[0 * NPIX + idx - 1];  d1 = pm[1 * NPIX + idx - 1]; }
        if (iu >= 1) { r0 = pm[0 * NPIX + idx - WH]; r1 = pm[1 * NPIX + idx - WH]; }
        float ab0 = p0 - r0, ab1 = p1 - r1;
        float bc0 = d0 - r0, bc1 = d1 - r1;
        float z = ab0 * bc1 - ab1 * bc0;
        float vis = (z > 0.f) ? 1.f : ((z < 0.f) ? 0.1f : z);
        // attention sample (JAX-style clamped gather)
        int ax = ((int)dst[0]) / 8;
        int ay = ((int)dst[1]) / 8;
        ax = ax < 0 ? 0 : (ax > 31 ? 31 : ax);
        ay = ay < 0 ? 0 : (ay > 31 ? 31 : ay);
        float wat = attention[(size_t)b * 1024 + ay * 32 + ax];
        s_w[k] = vis * wat;
    }
    __syncthreads();
    if (k != 0) return;

    // ---- serial weighted Procrustes for this batch ----
    float sw = 0.f, cs[3] = {0.f, 0.f, 0.f}, cd[3] = {0.f, 0.f, 0.f};
    for (int i = 0; i < NKPT; ++i) {
        float w = s_w[i]; sw += w;
        #pragma unroll
        for (int c = 0; c < 3; ++c) { cs[c] += w * s_src[i][c]; cd[c] += w * s_dst[i][c]; }
    }
    #pragma unroll
    for (int c = 0; c < 3; ++c) { cs[c] /= sw; cd[c] /= sw; }

    float sd1 = 0.f, sd2 = 0.f;
    for (int i = 0; i < NKPT; ++i) {
        float a0 = s_src[i][0] - cs[0], a1 = s_src[i][1] - cs[1], a2 = s_src[i][2] - cs[2];
        sd1 += sqrtf(a0 * a0 + a1 * a1 + a2 * a2);
        float b0 = s_dst[i][0] - cd[0], b1 = s_dst[i][1] - cd[1], b2 = s_dst[i][2] - cd[2];
        sd2 += sqrtf(b0 * b0 + b1 * b1 + b2 * b2);
    }
    float scale = sd2 / sd1;

    float mA[3] = {0.f, 0.f, 0.f}, mB[3] = {0.f, 0.f, 0.f};
    for (int i = 0; i < NKPT; ++i)
        #pragma unroll
        for (int c = 0; c < 3; ++c) { mA[c] += s_src[i][c] * scale; mB[c] += s_dst[i][c]; }
    #pragma unroll
    for (int c = 0; c < 3; ++c) { mA[c] *= (1.f / NKPT); mB[c] *= (1.f / NKPT); }

    float Hm[3][3] = {{0.f,0.f,0.f},{0.f,0.f,0.f},{0.f,0.f,0.f}};
    for (int i = 0; i < NKPT; ++i) {
        float w = s_w[i];
        float AA[3], BB[3];
        #pragma unroll
        for (int c = 0; c < 3; ++c) {
            AA[c] = s_src[i][c] * scale - mA[c];
            BB[c] = s_dst[i][c] - mB[c];
        }
        #pragma unroll
        for (int c = 0; c < 3; ++c)
            #pragma unroll
            for (int d = 0; d < 3; ++d)
                Hm[c][d] += w * AA[c] * BB[d];
    }

    // S = H^T H (symmetric), Jacobi eigendecomposition S = V diag(ev) V^T
    float S[3][3];
    #pragma unroll
    for (int i = 0; i < 3; ++i)
        #pragma unroll
        for (int j = 0; j < 3; ++j) {
            float s = 0.f;
            #pragma unroll
            for (int c = 0; c < 3; ++c) s += Hm[c][i] * Hm[c][j];
            S[i][j] = s;
        }
    float V[3][3] = {{1.f,0.f,0.f},{0.f,1.f,0.f},{0.f,0.f,1.f}};
    for (int sweep = 0; sweep < 16; ++sweep) {
        #pragma unroll
        for (int pair = 0; pair < 3; ++pair) {
            int p = (pair == 2) ? 1 : 0;
            int q = (pair == 0) ? 1 : 2;
            float apq = S[p][q];
            if (fabsf(apq) < 1e-20f) continue;
            float tau = (S[q][q] - S[p][p]) / (2.f * apq);
            float tt  = ((tau >= 0.f) ? 1.f : -1.f) / (fabsf(tau) + sqrtf(1.f + tau * tau));
            float cth = 1.f / sqrtf(1.f + tt * tt);
            float sth = tt * cth;
            for (int i = 0; i < 3; ++i) {
                float sp = S[i][p], sq = S[i][q];
                S[i][p] = cth * sp - sth * sq;
                S[i][q] = sth * sp + cth * sq;
            }
            for (int i = 0; i < 3; ++i) {
                float sp = S[p][i], sq = S[q][i];
                S[p][i] = cth * sp - sth * sq;
                S[q][i] = sth * sp + cth * sq;
            }
            for (int i = 0; i < 3; ++i) {
                float vp = V[i][p], vq = V[i][q];
                V[i][p] = cth * vp - sth * vq;
                V[i][q] = sth * vp + cth * vq;
            }
        }
    }
    float invs[3];
    #pragma unroll
    for (int m = 0; m < 3; ++m) {
        float sv = sqrtf(fmaxf(S[m][m], 0.f));
        invs[m] = (sv > 0.f) ? (1.f / sv) : 0.f;
    }
    // R = V diag(1/s) V^T H^T   (sign/order invariant form of V U^T)
    float P[3][3];
    #pragma unroll
    for (int i = 0; i < 3; ++i)
        #pragma unroll
        for (int j = 0; j < 3; ++j) {
            float s = 0.f;
            #pragma unroll
            for (int m = 0; m < 3; ++m) s += V[i][m] * V[j][m] * invs[m];
            P[i][j] = s;
        }
    float R[3][3];
    #pragma unroll
    for (int i = 0; i < 3; ++i)
        #pragma unroll
        for (int j = 0; j < 3; ++j) {
            float s = 0.f;
            #pragma unroll
            for (int m = 0; m < 3; ++m) s += P[i][m] * Hm[j][m];
            R[i][j] = s;
        }
    // t = mean_k( dst - (src*scale) R^T )
    float t[3] = {0.f, 0.f, 0.f};
    for (int i = 0; i < NKPT; ++i) {
        #pragma unroll
        for (int d = 0; d < 3; ++d) {
            float acc = s_dst[i][d];
            #pragma unroll
            for (int c = 0; c < 3; ++c) acc -= s_src[i][c] * scale * R[d][c];
            t[d] += acc;
        }
    }
    float* M = Mout + b * 12;
    #pragma unroll
    for (int d = 0; d < 3; ++d) {
        #pragma unroll
        for (int c = 0; c < 3; ++c) M[d * 3 + c] = R[d][c] * scale;   // Rs
        M[9 + d] = t[d] * (1.f / NKPT);
    }
}

// ---------------------------------------------------------------------------
// Phase B: out[b,d,w,h] = sum_c (Offset[b,c,p] + mean[c,p]) * Rs[b][d][c] + T[b][d]
// via V_WMMA_F32_16X16X4_F32: 16 pixels per WMMA tile, homogeneous K=4.
//   A (16x4): lanes 0-15 K=(x,y), lanes 16-31 K=(z,1)
//   B (4x16): lanes 0-15 n=lane -> (Rs[n][0],Rs[n][1]); lanes 16-31 -> (Rs[n][2],T[n])
//   D (16x16): lane n<3 holds channel n of pixels 0..7 (lanes 16..18: pixels 8..15)
// Each wave owns TILES_PER_WAVE contiguous tiles (128 contiguous pixels).
// Offset loads + output stores are non-temporal (single-use streams);
// mean_posmap (re-read by every batch) stays cache-resident.
// ---------------------------------------------------------------------------
__global__ __launch_bounds__(TB_THREADS) void vrm_transform_kernel(
    const float* __restrict__ Offset,
    const float* __restrict__ mean_posmap,
    const float* __restrict__ M,
    float* __restrict__ out)
{
    const int b    = blockIdx.y;
    const int lane = threadIdx.x & 31;
    const int n    = lane & 15;
    const int half = lane >> 4;
    const int waveId = blockIdx.x * (TB_THREADS / 32) + (threadIdx.x >> 5);

    // Build B operand (per-lane, once per wave)
    const float* Mb = M + b * 12;
    v2f bmat; bmat.x = 0.f; bmat.y = 0.f;
    if (n < 3) {
        if (half == 0) { bmat.x = Mb[n * 3 + 0]; bmat.y = Mb[n * 3 + 1]; }
        else           { bmat.x = Mb[n * 3 + 2]; bmat.y = Mb[9 + n];     }
    }

    const float* offB = Offset + (size_t)b * 3 * NPIX;
    float*       outB = out    + (size_t)b * 3 * NPIX;
    const int plane0 = half ? 2 : 0;  // K0 source plane for this half-wave

    const int t0 = waveId * TILES_PER_WAVE;
    #pragma unroll 4
    for (int i = 0; i < TILES_PER_WAVE; ++i) {
        const int t = t0 + i;
        const int p = t * 16 + n;

        // K0: plane 0 (low half) / plane 2 (high half)
        float a0 = __builtin_nontemporal_load(&offB[plane0 * NPIX + p])
                 + mean_posmap[plane0 * NPIX + p];
        // K1: plane 1 for low half; homogeneous 1.0 for high half.
        // High-half lanes hit the same cacheline as the low half -> free.
        float a1l = __builtin_nontemporal_load(&offB[NPIX + p])
                  + mean_posmap[NPIX + p];
        float a1 = half ? 1.0f : a1l;

        v2f a; a.x = a0; a.y = a1;
        v8f c = {};
        c = __builtin_amdgcn_wmma_f32_16x16x4_f32(
                /*neg_a=*/false, a, /*neg_b=*/false, bmat,
                /*c_mod=*/(short)0, c, /*reuse_a=*/false, /*reuse_b=*/false);

        if (n < 3) {
            float* dst = outB + (size_t)n * NPIX + t * 16 + half * 8;
            v4f lo, hi4;
            lo.x  = c[0]; lo.y  = c[1]; lo.z  = c[2]; lo.w  = c[3];
            hi4.x = c[4]; hi4.y = c[5]; hi4.z = c[6]; hi4.w = c[7];
            __builtin_nontemporal_store(lo,  (v4f*)(dst));
            __builtin_nontemporal_store(hi4, (v4f*)(dst + 4));
        }
    }
}

// ---------------------------------------------------------------------------
extern "C" void kernel_launch(void* const* d_in, const int* in_sizes, int n_in,
                              void* d_out, int out_size, void* d_ws, size_t ws_size,
                              hipStream_t stream) {
    const float* Offset      = (const float*)d_in[0];
    const float* Posmap_kpt  = (const float*)d_in[1];
    const float* attention   = (const float*)d_in[2];
    const float* mean_posmap = (const float*)d_in[3];
    const int*   uv_kpt_ind  = (const int*)d_in[4];
    float*       out         = (float*)d_out;
    float*       Mws         = (float*)d_ws;   // 64 * 12 floats

    vrm_procrustes_kernel<<<64, 128, 0, stream>>>(
        Offset, Posmap_kpt, attention, mean_posmap, uv_kpt_ind, Mws);

    // 64 blocks/batch * 8 waves = 512 waves per batch; 4096 tiles/batch
    // -> exactly TILES_PER_WAVE = 8 contiguous tiles per wave.
    vrm_transform_kernel<<<dim3(TB_BLOCKS_PER_BATCH, 64), TB_THREADS, 0, stream>>>(
        Offset, mean_posmap, Mws, out);
}